// ForgettingAwareReplay_38620345926151
// MI455X (gfx1250) — compile-verified
//
#include <hip/hip_runtime.h>
#include <hip/hip_bf16.h>

typedef float v2f __attribute__((ext_vector_type(2)));
typedef float v8f __attribute__((ext_vector_type(8)));

#define IOU_THRESH 0.5f

#if __has_builtin(__builtin_amdgcn_wmma_f32_16x16x4_f32)
#define HAVE_WMMA4 1
#else
#define HAVE_WMMA4 0
#endif

// ---------------------------------------------------------------------------
// Phase 1: fill IoU matrix [nGT x nRec]. One wave computes one 16x16 tile.
// The (area_a[M] + area_b[N]) outer-sum tile is produced by
// v_wmma_f32_16x16x4_f32:  A(16x4) = [areaA, 1, 0, 0] rows,
//                          B(4x16) = [1; areaB; 0; 0] cols.
// C/D layout: VGPR v, lanes 0-15 -> M=v, N=lane; lanes 16-31 -> M=v+8.
// ---------------------------------------------------------------------------
__global__ __launch_bounds__(256) void iou_tile_kernel(
    const float4* __restrict__ gt_boxes,
    const float4* __restrict__ rec_boxes,
    float* __restrict__ out, int tilesX, int nRec)
{
    const int lane  = threadIdx.x & 31;
    const int gwave = blockIdx.x * (blockDim.x >> 5) + (threadIdx.x >> 5);
    const int ty    = gwave / tilesX;
    const int tx    = gwave - ty * tilesX;
    const int mbase = ty << 4;
    const int col   = (tx << 4) + (lane & 15);

    // Record (column) box for this lane's N.
    const float4 rb = rec_boxes[col];
    const float areaB = (rb.z - rb.x) * (rb.w - rb.y);

    // GT (row) box for A-operand setup (lanes 0-15 carry K=0..1, lanes 16-31 K=2..3 = 0).
    const float4 gbl = gt_boxes[mbase + (lane & 15)];
    const float areaA = (gbl.z - gbl.x) * (gbl.w - gbl.y);

    const bool lo = lane < 16;
    v2f a = { lo ? areaA : 0.0f, lo ? 1.0f  : 0.0f };
    v2f b = { lo ? 1.0f  : 0.0f, lo ? areaB : 0.0f };
    v8f c = { 0.0f, 0.0f, 0.0f, 0.0f, 0.0f, 0.0f, 0.0f, 0.0f };
#if HAVE_WMMA4
    // D[M][N] = areaA[M]*1 + 1*areaB[N]
    c = __builtin_amdgcn_wmma_f32_16x16x4_f32(
        /*neg_a=*/false, a, /*neg_b=*/false, b,
        /*c_mod=*/(short)0, c, /*reuse_a=*/false, /*reuse_b=*/false);
#else
    (void)a; (void)b;
#endif

    const int half = (lane >> 4) << 3;  // 0 for lanes 0-15, 8 for lanes 16-31
#pragma unroll
    for (int v = 0; v < 8; ++v) {
        const int M = mbase + half + v;
        const float4 gb = gt_boxes[M];
        const float ltx = fmaxf(gb.x, rb.x), lty = fmaxf(gb.y, rb.y);
        const float rbx = fminf(gb.z, rb.z), rby = fminf(gb.w, rb.w);
        const float w = fmaxf(rbx - ltx, 0.0f), h = fmaxf(rby - lty, 0.0f);
        const float inter = w * h;
#if HAVE_WMMA4
        const float sums = c[v];
#else
        const float areaAM = (gb.z - gb.x) * (gb.w - gb.y);
        const float sums = areaAM + areaB;
#endif
        out[(size_t)M * nRec + col] = inter / (sums - inter);
    }
}

// ---------------------------------------------------------------------------
// Phase 2: sequential greedy matching. Single 1024-thread workgroup.
// claimed flags + record classes (+ record boxes in RECOMPUTE mode) live in
// LDS (CDNA5 WGP has 320KB). Each row: masked argmax over nRec records via
// thread-local scan + wave32 shuffle reduce + cross-wave LDS reduce.
// ---------------------------------------------------------------------------
template <bool RECOMPUTE>
__global__ __launch_bounds__(1024) void greedy_match_kernel(
    const float* __restrict__ iou,
    const float4* __restrict__ gt_boxes,
    const int* __restrict__ gt_labels,
    const float4* __restrict__ rec_boxes,
    const int* __restrict__ rec_classes,
    int nGT, int nRec,
    int* __restrict__ out_match, float* __restrict__ out_iou)
{
    extern __shared__ char smem[];
    int*    cls     = (int*)smem;                 // nRec ints
    int*    claimed = cls + nRec;                 // nRec ints
    float*  red_val = (float*)(claimed + nRec);   // 32 floats
    int*    red_idx = (int*)(red_val + 32);       // 32 ints
    float4* rbox    = (float4*)(red_idx + 32);    // nRec float4 (RECOMPUTE only)

    const int tid  = threadIdx.x;
    const int lane = tid & 31;
    const int wave = tid >> 5;
    const int nWaves = blockDim.x >> 5;

    for (int j = tid; j < nRec; j += blockDim.x) {
        cls[j]     = rec_classes[j];
        claimed[j] = 0;
        if (RECOMPUTE) rbox[j] = rec_boxes[j];
    }
    __syncthreads();

    const int per = nRec / blockDim.x;

    for (int r = 0; r < nGT; ++r) {
        const int lbl = gt_labels[r];
        const float* row = iou + (size_t)r * nRec;

        float4 gb;
        float areaA = 0.0f;
        if (RECOMPUTE) {
            gb = gt_boxes[r];
            areaA = (gb.z - gb.x) * (gb.w - gb.y);
        }

        float best = -2.0f;
        int   bidx = 0x7FFFFFFF;
        for (int k = 0; k < per; ++k) {
            const int j = tid + k * (int)blockDim.x;  // ascending j per thread
            float val = -1.0f;                        // masked value
            if (cls[j] == lbl && claimed[j] == 0) {
                if (RECOMPUTE) {
                    const float4 rb = rbox[j];
                    const float ltx = fmaxf(gb.x, rb.x), lty = fmaxf(gb.y, rb.y);
                    const float rbx = fminf(gb.z, rb.z), rby = fminf(gb.w, rb.w);
                    const float w = fmaxf(rbx - ltx, 0.0f), h = fmaxf(rby - lty, 0.0f);
                    const float inter = w * h;
                    const float areaB = (rb.z - rb.x) * (rb.w - rb.y);
                    val = inter / (areaA + areaB - inter);
                } else {
                    val = row[j];
                }
            }
            if (val > best || (val == best && j < bidx)) { best = val; bidx = j; }
        }

        // wave32 reduce (argmax, ties -> lowest index, matching jnp.argmax)
        for (int off = 16; off > 0; off >>= 1) {
            const float ov = __shfl_down(best, off, 32);
            const int   oi = __shfl_down(bidx, off, 32);
            if (ov > best || (ov == best && oi < bidx)) { best = ov; bidx = oi; }
        }
        if (lane == 0) { red_val[wave] = best; red_idx[wave] = bidx; }
        __syncthreads();

        if (wave == 0) {
            float bv = (lane < nWaves) ? red_val[lane] : -2.0f;
            int   bi = (lane < nWaves) ? red_idx[lane] : 0x7FFFFFFF;
            for (int off = 16; off > 0; off >>= 1) {
                const float ov = __shfl_down(bv, off, 32);
                const int   oi = __shfl_down(bi, off, 32);
                if (ov > bv || (ov == bv && oi < bi)) { bv = ov; bi = oi; }
            }
            if (lane == 0) {
                const bool take = bv >= IOU_THRESH;
                if (take) claimed[bi] = 1;
                out_match[r] = take ? bi : -1;
                out_iou[r]   = take ? bv : 0.0f;
            }
        }
        __syncthreads();  // claimed update visible before next row

        if (!RECOMPUTE && (r + 1 < nGT)) {
            // pull next row toward the WGP while this row's tail finishes
            __builtin_prefetch(row + nRec + tid, 0, 0);
        }
    }
}

// ---------------------------------------------------------------------------
extern "C" void kernel_launch(void* const* d_in, const int* in_sizes, int n_in,
                              void* d_out, int out_size, void* d_ws, size_t ws_size,
                              hipStream_t stream)
{
    const float4* gt_boxes    = (const float4*)d_in[0];   // [nGT,4] f32
    const int*    gt_labels   = (const int*)d_in[1];      // [nGT]   i32
    const float4* rec_boxes   = (const float4*)d_in[2];   // [nRec,4] f32
    const int*    rec_classes = (const int*)d_in[3];      // [nRec]  i32

    const int nGT  = in_sizes[1];
    const int nRec = in_sizes[3];

    int*   out_match = (int*)d_out;            // matches: int32[nGT]
    float* out_iou   = (float*)d_out + nGT;    // matched_ious: f32[nGT]

    const size_t iou_bytes = (size_t)nGT * (size_t)nRec * sizeof(float);
    const size_t sh_base   = (size_t)nRec * sizeof(int) * 2 + 64 * sizeof(int);

    if (ws_size >= iou_bytes) {
        float* iou = (float*)d_ws;
        const int tilesX = nRec / 16;
        const int tilesY = nGT / 16;
        const int waves  = tilesX * tilesY;       // one tile per wave
        const int blocks = waves / 8;             // 8 waves per 256-thread block
        iou_tile_kernel<<<blocks, 256, 0, stream>>>(gt_boxes, rec_boxes, iou,
                                                    tilesX, nRec);
        greedy_match_kernel<false><<<1, 1024, sh_base, stream>>>(
            iou, gt_boxes, gt_labels, rec_boxes, rec_classes,
            nGT, nRec, out_match, out_iou);
    } else {
        // Workspace too small for the IoU matrix: recompute IoUs on the fly
        // from LDS-resident record boxes (nRec*16B extra LDS, fits in 320KB).
        const size_t sh = sh_base + (size_t)nRec * sizeof(float4);
        greedy_match_kernel<true><<<1, 1024, sh, stream>>>(
            nullptr, gt_boxes, gt_labels, rec_boxes, rec_classes,
            nGT, nRec, out_match, out_iou);
    }
}